// My_Loss_eval_60327110639931
// MI455X (gfx1250) — compile-verified
//
#include <hip/hip_runtime.h>

// ---------------------------------------------------------------------------
// My_Loss_eval: cls NLL + pairwise hash loss on MI455X (gfx1250).
//
// hash part: theta = H H^T / 2  (8192x8192x64 fp32 GEMM, 8.6 GFLOP) fused with
// softplus epilogue + weighted reduction. theta is never materialized.
// Uses V_WMMA_F32_16X16X4_F32 (full fp32 precision; GEMM cost is dwarfed by
// the 134M transcendentals in the epilogue, so fp32 WMMA is the right pick).
// Symmetry exploited: only tiles touching the upper triangle are computed,
// off-diagonal elements weighted x2.
// ---------------------------------------------------------------------------

#define N_ROWS      8192
#define BITS        64
#define NUM_CLASSES 101
#define RT          64     // rows per block tile
#define CT          128    // cols per block tile
#define LDK         68     // padded LDS stride (272B: 16B aligned, bank-conflict free)

typedef __attribute__((ext_vector_type(2))) float v2f;
typedef __attribute__((ext_vector_type(4))) float v4f;
typedef __attribute__((ext_vector_type(8))) float v8f;

// ws layout (floats): [0]=pos_sum  [1]=neg_sum  [2]=sim_count  [3]=cls_sum
__global__ void init_ws_kernel(float* __restrict__ ws) {
    if (threadIdx.x < 4) ws[threadIdx.x] = 0.0f;
}

__launch_bounds__(256)
__global__ void hash_tile_kernel(const float* __restrict__ H,
                                 const int*   __restrict__ tgt,
                                 float*       __restrict__ ws) {
    const int bx = blockIdx.x;                 // col block: cols [bx*CT, bx*CT+CT)
    const int by = blockIdx.y;                 // row block: rows [by*RT, by*RT+RT)
    if (by >= 2 * (bx + 1)) return;            // block entirely below the diagonal

    __shared__ float As[RT * LDK];             // 17.0 KB
    __shared__ float Bs[CT * LDK];             // 34.0 KB
    __shared__ int   trs[RT];
    __shared__ int   tcs[CT];
    __shared__ float red[3];

    const int tid     = threadIdx.x;
    const int rowBase = by * RT;
    const int colBase = bx * CT;

    // Stage A tile (RT x 64 fp32) into LDS, v4f granularity.
    for (int q = tid; q < RT * (BITS / 4); q += 256) {
        const int r  = q >> 4;
        const int k4 = (q & 15) << 2;
        v4f v = *(const v4f*)(H + (size_t)(rowBase + r) * BITS + k4);
        *(v4f*)(As + r * LDK + k4) = v;
    }
    // Stage B tile (CT x 64 fp32).
    for (int q = tid; q < CT * (BITS / 4); q += 256) {
        const int r  = q >> 4;
        const int k4 = (q & 15) << 2;
        v4f v = *(const v4f*)(H + (size_t)(colBase + r) * BITS + k4);
        *(v4f*)(Bs + r * LDK + k4) = v;
    }
    if (tid < RT) trs[tid] = tgt[rowBase + tid];
    if (tid < CT) tcs[tid] = tgt[colBase + tid];
    if (tid < 3)  red[tid] = 0.0f;
    __syncthreads();

    const int wave  = tid >> 5;                // 0..7
    const int lane  = tid & 31;
    const int m0    = (wave & 3) * 16;         // row strip within A tile
    const int nbase = (wave >> 2) * 64;        // column half within B tile
    const int mrow  = lane & 15;               // M (for A) / N (for B) lane index
    const int hi    = lane >> 4;               // half-wave selects K pair

    // Hoist all A fragments for this strip: k0 = 4*kk, lane holds K = k0+2*hi, k0+2*hi+1.
    v2f areg[16];
#pragma unroll
    for (int kk = 0; kk < 16; ++kk)
        areg[kk] = *(const v2f*)(As + (m0 + mrow) * LDK + kk * 4 + 2 * hi);

    float pos_acc = 0.0f, neg_acc = 0.0f, cnt_acc = 0.0f;

#pragma unroll
    for (int s = 0; s < 4; ++s) {
        const int n0 = nbase + s * 16;
        v8f c = {};
#pragma unroll
        for (int kk = 0; kk < 16; ++kk) {
            v2f b = *(const v2f*)(Bs + (n0 + mrow) * LDK + kk * 4 + 2 * hi);
            // 8 args: (neg_a, A, neg_b, B, c_mod, C, reuse_a, reuse_b)
            c = __builtin_amdgcn_wmma_f32_16x16x4_f32(
                    false, areg[kk], false, b, (short)0, c, false, false);
        }
        // Epilogue: C layout — VGPR r holds M=r (lanes 0-15) / M=r+8 (lanes 16-31).
        const int gj  = colBase + n0 + mrow;
        const int tcv = tcs[n0 + mrow];
#pragma unroll
        for (int r = 0; r < 8; ++r) {
            const int mloc = m0 + r + hi * 8;
            const int gi   = rowBase + mloc;
            if (gi > gj) continue;                       // covered by symmetry
            const float sim = (trs[mloc] == tcv) ? 1.0f : 0.0f;
            if (gi == gj) { cnt_acc += 1.0f; continue; } // diag: counts for Sim.sum only
            cnt_acc += 2.0f * sim;
            const float theta = c[r] * 0.5f;
            const float sp    = __logf(1.0f + __expf(theta));   // log1p(exp(theta))
            if (sim != 0.0f) pos_acc += 2.0f * (sp - theta);
            else             neg_acc += 2.0f * sp;
        }
    }

    // wave32 shuffle reduction, then LDS, then one global atomic per block.
#pragma unroll
    for (int off = 16; off > 0; off >>= 1) {
        pos_acc += __shfl_down(pos_acc, off, 32);
        neg_acc += __shfl_down(neg_acc, off, 32);
        cnt_acc += __shfl_down(cnt_acc, off, 32);
    }
    if (lane == 0) {
        atomicAdd(&red[0], pos_acc);
        atomicAdd(&red[1], neg_acc);
        atomicAdd(&red[2], cnt_acc);
    }
    __syncthreads();
    if (tid == 0) {
        atomicAdd(&ws[0], red[0]);
        atomicAdd(&ws[1], red[1]);
        atomicAdd(&ws[2], red[2]);
    }
}

__launch_bounds__(256)
__global__ void cls_kernel(const float* __restrict__ X,
                           const int*   __restrict__ tgt,
                           float*       __restrict__ ws) {
    const int row  = blockIdx.x * 8 + (threadIdx.x >> 5);   // one wave32 per row
    const int lane = threadIdx.x & 31;
    const float* x = X + (size_t)row * NUM_CLASSES;

    float m = -3.4e38f;
    for (int cc = lane; cc < NUM_CLASSES; cc += 32) m = fmaxf(m, x[cc]);
#pragma unroll
    for (int off = 16; off > 0; off >>= 1) m = fmaxf(m, __shfl_xor(m, off, 32));

    float se = 0.0f;
    for (int cc = lane; cc < NUM_CLASSES; cc += 32) se += __expf(x[cc] - m);
#pragma unroll
    for (int off = 16; off > 0; off >>= 1) se += __shfl_xor(se, off, 32);

    if (lane == 0) {
        const float lse = m + __logf(se);
        const float nll = lse - x[tgt[row]];
        atomicAdd(&ws[3], nll);
    }
}

__global__ void finalize_kernel(const float* __restrict__ ws,
                                float*       __restrict__ out) {
    if (threadIdx.x != 0 || blockIdx.x != 0) return;
    const double Nn    = (double)N_ROWS;
    const double n_pos = (double)ws[2];
    double S1 = n_pos - Nn;
    double S0 = Nn * Nn - n_pos;
    if (S0 == 0.0) S0 = 1.0;
    if (S1 == 0.0) S1 = 1.0;
    const double S     = S0 + S1;
    const double hsum  = (double)ws[0] * (S / S1) + (double)ws[1] * (S / S0);
    const double cnt   = Nn * (Nn - 1.0) * 0.5;
    const double hash_loss = hsum * 0.5 / cnt;
    const double cls_loss  = (double)ws[3] / Nn;
    out[0] = (float)hash_loss;
    out[1] = (float)cls_loss;
    out[2] = (float)(1.0 * cls_loss + 0.01 * hash_loss);   // GAMM=1.0, ALPH=0.01
}

extern "C" void kernel_launch(void* const* d_in, const int* in_sizes, int n_in,
                              void* d_out, int out_size, void* d_ws, size_t ws_size,
                              hipStream_t stream) {
    const float* hash_out = (const float*)d_in[0];   // 8192 x 64 fp32
    const float* cls_out  = (const float*)d_in[1];   // 8192 x 101 fp32
    const int*   target   = (const int*)d_in[2];     // 8192 ints
    float* out = (float*)d_out;                      // 3 fp32 scalars
    float* ws  = (float*)d_ws;

    init_ws_kernel<<<1, 32, 0, stream>>>(ws);

    dim3 grid(N_ROWS / CT, N_ROWS / RT);             // (64, 128); lower-tri blocks exit
    hash_tile_kernel<<<grid, 256, 0, stream>>>(hash_out, target, ws);

    cls_kernel<<<N_ROWS / 8, 256, 0, stream>>>(cls_out, target, ws);

    finalize_kernel<<<1, 1, 0, stream>>>(ws, out);
}